// VSSM_V_29265907155741
// MI455X (gfx1250) — compile-verified
//
#include <hip/hip_runtime.h>
#include <hip/hip_bf16.h>

typedef __attribute__((ext_vector_type(2))) float v2f;
typedef __attribute__((ext_vector_type(8))) float v8f;

#define B_SZ   4
#define HW_    1024
#define C_IN   96
#define DINNER 192
#define EPROJ  384      // 2*DINNER
#define NST    16
#define DTRANK 512
#define CPROJ  544      // DTRANK + 2*NST
#define NPIX   4096     // B_SZ*HW_

// ---------------------------------------------------------------------------
// fp32 WMMA GEMM: one wave32 computes a 16 x (16*NT) strip of C, reusing the
// A fragment across NT column tiles (NT v_wmma per A-fragment load).
//   C[M x N] = A[M x K] * B            (row-major C, ldc)
//   BT=true : B row-major [N x K] (ldb = K-stride)  -> C = A * B^T
//   BT=false: B row-major [K x N] (ldb = N-stride)  -> C = A * B
// Batch over blockIdx.z: A advances by aStride per slot, wrapping every aPeriod slots,
// B by bStride, C by cStride.
// Fragment layout per ISA 7.12.2 (f32, wave32):
//   A 16x4 : lane row = lane&15, ks = {2*(lane>>4), 2*(lane>>4)+1}
//   B 4x16 : lane col = lane&15, same k split
//   C 16x16: vgpr i -> row (i + 8*(lane>>4)), col lane&15
// ---------------------------------------------------------------------------
template <bool BT, int NT>
__global__ void wmma_gemm_f32(const float* __restrict__ A,
                              const float* __restrict__ Bm,
                              float* __restrict__ C,
                              int K, int lda, int ldb, int ldc,
                              long aStride, int aPeriod,
                              long bStride, long cStride) {
    const int z = (int)blockIdx.z;
    const float* Ap = A  + (long)(z % aPeriod) * aStride;
    const float* Bp = Bm + (long)z * bStride;
    float*       Cp = C  + (long)z * cStride;

    const int lane = (int)threadIdx.x;      // 0..31, full wave (EXEC all ones)
    const int row  = lane & 15;
    const int half = lane >> 4;             // 0 or 1
    const int m0   = (int)blockIdx.x * 16;
    const int n0   = (int)blockIdx.y * (16 * NT);

    v8f acc[NT];
#pragma unroll
    for (int j = 0; j < NT; ++j) acc[j] = (v8f){};

    const float* arow = Ap + (size_t)(m0 + row) * lda;

    for (int k0 = 0; k0 < K; k0 += 4) {
        const int ka = k0 + 2 * half;
        v2f a;
        a.x = arow[ka];
        a.y = arow[ka + 1];
#pragma unroll
        for (int j = 0; j < NT; ++j) {
            v2f b;
            if (BT) {
                const float* brow = Bp + (size_t)(n0 + 16 * j + row) * ldb;
                b.x = brow[ka];
                b.y = brow[ka + 1];
            } else {
                b.x = Bp[(size_t)ka       * ldb + n0 + 16 * j + row];
                b.y = Bp[(size_t)(ka + 1) * ldb + n0 + 16 * j + row];
            }
            acc[j] = __builtin_amdgcn_wmma_f32_16x16x4_f32(
                /*neg_a=*/false, a, /*neg_b=*/false, b,
                /*c_mod=*/(short)0, acc[j], /*reuse_a=*/false, /*reuse_b=*/false);
        }
    }
#pragma unroll
    for (int j = 0; j < NT; ++j)
#pragma unroll
        for (int i = 0; i < 8; ++i)
            Cp[(size_t)(m0 + i + 8 * half) * ldc + (n0 + 16 * j + row)] = acc[j][i];
}

// ---------------------------------------------------------------------------
// 3x3 SAME conv (NCHW math, data held NHWC) + bias + SiLU.
// One block per pixel (b,h,w); 192 threads = output channels.
// Writes xs[b,0,hw,l] = v  and  xs[b,1,hw,191-l] = v (channel-flipped copy).
// ---------------------------------------------------------------------------
__global__ void conv3x3_silu_kernel(const float* __restrict__ xz,      // [4096 x 384]
                                    const float* __restrict__ conv_w,  // [192][192][3][3]
                                    const float* __restrict__ conv_b,  // [192]
                                    float* __restrict__ xs) {          // [B,2,HW,192]
    __shared__ float xn[9 * DINNER];
    const int co = (int)threadIdx.x;           // 0..191
    const int pix = (int)blockIdx.x;           // 0..4095
    const int b  = pix >> 10;
    const int hw = pix & 1023;
    const int h  = hw >> 5, w = hw & 31;

#pragma unroll
    for (int t = 0; t < 9; ++t) {
        const int hh = h + t / 3 - 1;
        const int ww = w + t % 3 - 1;
        float v = 0.f;
        if (hh >= 0 && hh < 32 && ww >= 0 && ww < 32)
            v = xz[(size_t)(b * HW_ + hh * 32 + ww) * EPROJ + co];
        xn[t * DINNER + co] = v;
    }
    __syncthreads();

    float acc = conv_b[co];
    const float* wrow = conv_w + (size_t)co * DINNER * 9;
    for (int ci = 0; ci < DINNER; ++ci) {
        const float* wt = wrow + ci * 9;
#pragma unroll
        for (int t = 0; t < 9; ++t)
            acc = fmaf(xn[t * DINNER + ci], wt[t], acc);
    }
    const float val = acc / (1.f + __expf(-acc));   // SiLU
    const size_t fwd = ((size_t)(b * 2 + 0) * HW_ + hw) * DINNER;
    const size_t bwd = ((size_t)(b * 2 + 1) * HW_ + hw) * DINNER;
    xs[fwd + co]                = val;
    xs[bwd + (DINNER - 1 - co)] = val;
}

// ---------------------------------------------------------------------------
// Selective scan. One thread per (b,k,d); loop l = 0..191, 16 states in regs.
// B/C projections (xdbl rows 512..543 = one contiguous 24 KB block) are staged
// into LDS with GLOBAL_LOAD_ASYNC_TO_LDS_B128 (ASYNCcnt-tracked, no VGPR
// round-trip), overlapped with per-thread A/bias loads.
// ---------------------------------------------------------------------------
__global__ void scan_kernel(const float* __restrict__ xdbl,   // [8,544,192]
                            const float* __restrict__ dtf,    // [8,1024,192]
                            const float* __restrict__ xs,     // [8,1024,192]
                            const float* __restrict__ A_logs, // [2048,16]
                            const float* __restrict__ Ds,     // [2048]
                            const float* __restrict__ dtb,    // [2,1024]
                            float* __restrict__ ysc) {        // [8,1024,192]
    __shared__ __attribute__((aligned(16))) float BC[2 * NST * DINNER]; // Bl | Cl

    const int pair = (int)blockIdx.y;                 // 0..7 = (b,k)
    const int k    = pair & 1;
    const int tid  = (int)threadIdx.x;
    const int d    = (int)blockIdx.x * 256 + tid;     // 0..1023

    // --- async bulk copy: 6144 floats, 256 threads x 6 x b128 ---
    {
        const float* src = xdbl + (size_t)pair * CPROJ * DINNER
                                + (size_t)DTRANK * DINNER;
        const unsigned ldsBase = (unsigned)(size_t)(&BC[0]); // flat low 32 = LDS offset
#pragma unroll
        for (int j = 0; j < 6; ++j) {
            const int e = (j * 256 + tid) * 4;               // float index, 16B chunks
            const unsigned      ldsA = ldsBase + (unsigned)(e * 4);
            const unsigned long long g = (unsigned long long)(const void*)(src + e);
            asm volatile("global_load_async_to_lds_b128 %0, %1, off"
                         :: "v"(ldsA), "v"(g) : "memory");
        }
    }

    // per-thread parameters (overlap with async copy in flight)
    const int kd = k * HW_ + d;
    float An[NST];
#pragma unroll
    for (int n = 0; n < NST; ++n)
        An[n] = -__expf(A_logs[(size_t)kd * NST + n]);
    const float bias = dtb[kd];
    const float Dsv  = Ds[kd];

    asm volatile("s_wait_asynccnt 0x0" ::: "memory");
    __syncthreads();

    const float* Bl = BC;
    const float* Cl = BC + NST * DINNER;

    float h[NST];
#pragma unroll
    for (int n = 0; n < NST; ++n) h[n] = 0.f;

    const size_t rowbase = ((size_t)pair * HW_ + d) * DINNER;
    for (int l = 0; l < DINNER; ++l) {
        const float xv = dtf[rowbase + l] + bias;
        const float delta = (xv > 20.f) ? xv : log1pf(__expf(xv));  // softplus
        const float u  = xs[rowbase + l];
        const float du = delta * u;
        float y = 0.f;
#pragma unroll
        for (int n = 0; n < NST; ++n) {
            h[n] = __expf(delta * An[n]) * h[n] + du * Bl[n * DINNER + l];
            y = fmaf(h[n], Cl[n * DINNER + l], y);
        }
        ysc[rowbase + l] = y + Dsv * u;
    }
}

// ---------------------------------------------------------------------------
// Combine fwd + flipped-bwd, LayerNorm over d=1024 per (b,l), gate by SiLU(z).
// One block per (l,b); 256 threads.
// ---------------------------------------------------------------------------
__global__ void ln_gate_kernel(const float* __restrict__ ysc,   // [8,1024,192]
                               const float* __restrict__ xz,    // [4096,384]
                               const float* __restrict__ gamma, // [1024]
                               const float* __restrict__ beta,  // [1024]
                               float* __restrict__ ysil) {      // [4096,192]
    __shared__ float ybuf[HW_];
    __shared__ float red[256];
    __shared__ float red2[256];

    const int l   = (int)blockIdx.x;     // 0..191
    const int b   = (int)blockIdx.y;     // 0..3
    const int tid = (int)threadIdx.x;

    float s = 0.f, s2 = 0.f;
    for (int d = tid; d < HW_; d += 256) {
        const float v =
            ysc[((size_t)(b * 2 + 0) * HW_ + d) * DINNER + l] +
            ysc[((size_t)(b * 2 + 1) * HW_ + d) * DINNER + (DINNER - 1 - l)];
        ybuf[d] = v;
        s  += v;
        s2 += v * v;
    }
    red[tid] = s; red2[tid] = s2;
    __syncthreads();
    for (int off = 128; off > 0; off >>= 1) {
        if (tid < off) { red[tid] += red[tid + off]; red2[tid] += red2[tid + off]; }
        __syncthreads();
    }
    const float mu   = red[0] * (1.f / HW_);
    const float var  = red2[0] * (1.f / HW_) - mu * mu;
    const float rstd = rsqrtf(var + 1e-5f);

    for (int d = tid; d < HW_; d += 256) {
        const float vn = (ybuf[d] - mu) * rstd * gamma[d] + beta[d];
        const float zv = xz[(size_t)(b * HW_ + d) * EPROJ + DINNER + l];
        const float sz = zv / (1.f + __expf(-zv));
        ysil[(size_t)(b * HW_ + d) * DINNER + l] = vn * sz;
    }
}

// ---------------------------------------------------------------------------
extern "C" void kernel_launch(void* const* d_in, const int* in_sizes, int n_in,
                              void* d_out, int out_size, void* d_ws, size_t ws_size,
                              hipStream_t stream) {
    (void)in_sizes; (void)n_in; (void)out_size; (void)ws_size;
    const float* x        = (const float*)d_in[0];   // [4,32,32,96]
    const float* in_w     = (const float*)d_in[1];   // [384,96]
    const float* conv_w   = (const float*)d_in[2];   // [192,192,3,3]
    const float* conv_b   = (const float*)d_in[3];   // [192]
    const float* xproj_w  = (const float*)d_in[4];   // [2,544,1024]
    const float* dtproj_w = (const float*)d_in[5];   // [2,1024,512]
    const float* dtb      = (const float*)d_in[6];   // [2,1024]
    const float* A_logs   = (const float*)d_in[7];   // [2048,16]
    const float* Ds       = (const float*)d_in[8];   // [2048]
    const float* gamma    = (const float*)d_in[9];   // [1024]
    const float* beta     = (const float*)d_in[10];  // [1024]
    const float* out_w    = (const float*)d_in[11];  // [96,192]
    float* out = (float*)d_out;                      // [4096,96]

    float* ws   = (float*)d_ws;
    float* xz   = ws;                                  // 4096*384
    float* xs   = xz   + (size_t)NPIX * EPROJ;         // 8*1024*192
    float* xdbl = xs   + (size_t)8 * HW_ * DINNER;     // 8*544*192
    float* dtf  = xdbl + (size_t)8 * CPROJ * DINNER;   // 8*1024*192
    float* ysc  = dtf  + (size_t)8 * HW_ * DINNER;     // 8*1024*192
    float* ysil = ysc  + (size_t)8 * HW_ * DINNER;     // 4096*192

    // 1) in_proj: xz[4096x384] = x[4096x96] * in_w^T   (NT=4 -> 16x64 per wave)
    wmma_gemm_f32<true, 4><<<dim3(NPIX / 16, EPROJ / 64, 1), dim3(32), 0, stream>>>(
        x, in_w, xz, C_IN, C_IN, C_IN, EPROJ, 0, 1, 0, 0);

    // 2) conv3x3 + bias + SiLU -> xs (fwd + channel-flipped)
    conv3x3_silu_kernel<<<dim3(NPIX), dim3(DINNER), 0, stream>>>(xz, conv_w, conv_b, xs);

    // 3) x_proj (8 batched): xdbl[544x192] = xproj_w[k] * xs[b,k]
    wmma_gemm_f32<false, 4><<<dim3(CPROJ / 16, DINNER / 64, 8), dim3(32), 0, stream>>>(
        xproj_w, xs, xdbl, HW_, HW_, DINNER, DINNER,
        (long)CPROJ * HW_, 2, (long)HW_ * DINNER, (long)CPROJ * DINNER);

    // 4) dt_proj (8 batched): dtf[1024x192] = dtproj_w[k] * xdbl[b,k][0:512]
    wmma_gemm_f32<false, 4><<<dim3(HW_ / 16, DINNER / 64, 8), dim3(32), 0, stream>>>(
        dtproj_w, xdbl, dtf, DTRANK, DTRANK, DINNER, DINNER,
        (long)HW_ * DTRANK, 2, (long)CPROJ * DINNER, (long)HW_ * DINNER);

    // 5) selective scan (async-to-LDS staging of B/C)
    scan_kernel<<<dim3(HW_ / 256, 8), dim3(256), 0, stream>>>(
        xdbl, dtf, xs, A_logs, Ds, dtb, ysc);

    // 6) combine + LayerNorm + SiLU(z) gating
    ln_gate_kernel<<<dim3(DINNER, B_SZ), dim3(256), 0, stream>>>(
        ysc, xz, gamma, beta, ysil);

    // 7) out_proj: out[4096x96] = ysil[4096x192] * out_w^T (NT=3 -> 16x48/wave)
    wmma_gemm_f32<true, 3><<<dim3(NPIX / 16, C_IN / 48, 1), dim3(32), 0, stream>>>(
        ysil, out_w, out, DINNER, DINNER, DINNER, C_IN, 0, 1, 0, 0);
}